// TensorProductCuda_65807488909771
// MI455X (gfx1250) — compile-verified
//
#include <hip/hip_runtime.h>

// Sparse CG tensor product densified to a per-node GEMM:
//   out[n] (156x64) = M[n] (156x16, from cg*y scatter) @ x[n] (16x64)
// computed with V_WMMA_F32_16X16X4_F32 (f32 in/out, full precision).

typedef __attribute__((ext_vector_type(2))) float v2f;
typedef __attribute__((ext_vector_type(8))) float v8f;

#define DIM_IN      16   // (LMAX+1)^2
#define NCH         64   // channels
#define DIM_OUT     156
#define DIM_OUT_PAD 160  // 10 tiles of 16 rows
#define BLOCK       128  // 4 wave32s; each wave owns a 16-channel slice

__global__ __launch_bounds__(BLOCK) void cg_tp_wmma_f32(
    const float* __restrict__ x,    // [N,16,64]
    const float* __restrict__ y,    // [N,16]
    const float* __restrict__ cg,   // [nnz]
    const int*   __restrict__ mu1,  // [nnz]
    const int*   __restrict__ mu2,  // [nnz]
    const int*   __restrict__ mu3,  // [nnz]
    float*       __restrict__ out,  // [N,156,64]
    int nnz)
{
    __shared__ float yl[DIM_IN];
    __shared__ float Ml[DIM_OUT_PAD * DIM_IN];  // 160x16 dense M, 10.0 KB

    const int n    = blockIdx.x;
    const int t    = threadIdx.x;
    const int lane = t & 31;
    const int wave = t >> 5;

    // ---- Phase 1: densify M[mu3,mu1] = cg * y[mu2] into LDS ----
    if (t < DIM_IN) yl[t] = y[(size_t)n * DIM_IN + t];
    #pragma unroll
    for (int i = t; i < DIM_OUT_PAD * DIM_IN; i += BLOCK) Ml[i] = 0.0f;
    __syncthreads();

    // Each (mu3,mu1) cell is written by at most one instruction -> no atomics.
    for (int i = t; i < nnz; i += BLOCK) {
        Ml[mu3[i] * DIM_IN + mu1[i]] = cg[i] * yl[mu2[i]];
    }
    __syncthreads();

    // ---- Phase 2: per-wave GEMM on a 16-channel slice ----
    // WMMA f32 16x16x4 operand layout (wave32):
    //   A (16x4): lanes 0-15 -> M=lane, v0=K0,v1=K1; lanes 16-31 -> M=lane-16, v0=K2,v1=K3
    //   B (4x16): lanes 0-15 -> N=lane, v0=K0,v1=K1; lanes 16-31 -> N=lane-16, v0=K2,v1=K3
    //   C/D (16x16): vgpr j -> M=j (lanes 0-15) / M=j+8 (lanes 16-31), N=lane%16
    const int ch   = (wave << 4) + (lane & 15);  // this lane's output channel
    const int khi  = (lane & 16) ? 2 : 0;        // K offset for upper lane half
    const int mlo  = lane & 15;                  // A-row within tile
    const int mhib = (lane & 16) ? 8 : 0;        // D-row base for this lane half

    // Load B fragments (x tile) once; reused across all 10 row tiles.
    const float* xb = x + (size_t)n * (DIM_IN * NCH) + ch;
    v2f b[4];
    #pragma unroll
    for (int kk = 0; kk < 4; ++kk) {
        const int kb = 4 * kk + khi;
        b[kk].x = xb[kb * NCH];
        b[kk].y = xb[(kb + 1) * NCH];
    }

    float* outp = out + (size_t)n * (DIM_OUT * NCH) + ch;

    #pragma unroll
    for (int r = 0; r < DIM_OUT_PAD / 16; ++r) {
        v8f acc = {};
        #pragma unroll
        for (int kk = 0; kk < 4; ++kk) {
            const int kb = 4 * kk + khi;
            v2f a;  // contiguous 8B pair in a 16-float LDS row -> ds_load_b64
            a.x = Ml[(r * 16 + mlo) * DIM_IN + kb];
            a.y = Ml[(r * 16 + mlo) * DIM_IN + kb + 1];
            acc = __builtin_amdgcn_wmma_f32_16x16x4_f32(
                /*neg_a=*/false, a, /*neg_b=*/false, b[kk],
                /*c_mod=*/(short)0, acc, /*reuse_a=*/false, /*reuse_b=*/false);
        }
        // Store 16x16 f32 D tile; guard trims the 4 pad rows (156..159).
        #pragma unroll
        for (int j = 0; j < 8; ++j) {
            const int row = r * 16 + mhib + j;
            if (row < DIM_OUT) outp[(size_t)row * NCH] = acc[j];
        }
    }
}

extern "C" void kernel_launch(void* const* d_in, const int* in_sizes, int n_in,
                              void* d_out, int out_size, void* d_ws, size_t ws_size,
                              hipStream_t stream) {
    const float* x   = (const float*)d_in[0];
    const float* y   = (const float*)d_in[1];
    const float* cg  = (const float*)d_in[2];
    const int*   mu1 = (const int*)d_in[3];
    const int*   mu2 = (const int*)d_in[4];
    const int*   mu3 = (const int*)d_in[5];
    float*       out = (float*)d_out;

    const int n_nodes = in_sizes[0] / (DIM_IN * NCH);  // 8192
    const int nnz     = in_sizes[2];                   // 562

    cg_tp_wmma_f32<<<n_nodes, BLOCK, 0, stream>>>(x, y, cg, mu1, mu2, mu3, out, nnz);
}